// Wavefront_SoS_73048803770441
// MI455X (gfx1250) — compile-verified
//
#include <hip/hip_runtime.h>

#define N_POINTS 1440
#define N_INT    4000
#define N_GRID   4096
#define BLOCK    256   // 8 wave32s per block

typedef float v2f __attribute__((ext_vector_type(2)));
typedef float v8f __attribute__((ext_vector_type(8)));

__global__ __launch_bounds__(BLOCK) void wavefront_sos_kernel(
    const float* __restrict__ xp,  const float* __restrict__ yp,
    const float* __restrict__ SoS, const float* __restrict__ thetas,
    const float* __restrict__ steps,
    const float* __restrict__ x0p, const float* __restrict__ y0p,
    const float* __restrict__ dxp, const float* __restrict__ dyp,
    const float* __restrict__ Rp,  const float* __restrict__ v0p,
    float* __restrict__ out)
{
    __shared__ float s_steps[N_INT];        // 16 KB of 320 KB/WGP: high occupancy OK
    __shared__ float s_red[BLOCK / 32];

    const int p   = blockIdx.x;             // one ray (theta) per block
    const int tid = threadIdx.x;

    // Stage the integration abscissae in LDS (read 3x per node otherwise).
    for (int i = tid; i < N_INT; i += BLOCK)
        s_steps[i] = steps[i];

    const float x  = xp[0],  y  = yp[0];
    const float x0 = x0p[0], y0 = y0p[0];
    const float dx = dxp[0], dy = dyp[0];
    const float Rb = Rp[0],  v0 = v0p[0];
    const float theta = thetas[p];

    // Chord length l for this angle (matches reference math, f32 throughout).
    // All of this is uniform per block -> lands on the scalar pipe.
    const float r    = sqrtf(x * x + y * y);
    const float phi  = atan2f(x, y);               // note: atan2(x, y) as in reference
    const float sd   = sinf(theta - phi);
    const float cd   = cosf(theta - phi);          // cos(phi-theta) == cos(theta-phi)
    const float s    = r * sd;
    const float arg  = fmaxf(Rb * Rb - s * s, 0.0f);
    const float sq   = sqrtf(arg);
    const float l_in  = sq + r * cd;
    const float l_out = 2.0f * sq * (cd >= 0.0f ? 1.0f : 0.0f);
    const float l    = (r < Rb) ? l_in : l_out;
    const float sth  = sinf(theta);
    const float cth  = cosf(theta);
    const float inv_dx = 1.0f / dx;                // prefetch-path only (approx OK)
    const float inv_dy = 1.0f / dy;

    __syncthreads();

    // Per-node trapezoid weight x value contribution; branch-free tail via
    // clamped index + zeroed weight. Prefetches its own chain 2 strides ahead.
    auto contrib = [&](int j) -> float {
        {   // prefetch next node on this accumulator chain (approx addr, rcp-mul)
            const int jn = min(j + 2 * BLOCK, N_INT - 1);
            const float xs2 = l * s_steps[jn];
            const float px2 = x - xs2 * sth;
            const float py2 = y - xs2 * cth;
            int xi2 = (int)rintf((px2 - x0) * inv_dx);
            int yi2 = (int)rintf((py2 - y0) * inv_dy);
            const int row2 = (-yi2) & (N_GRID - 1);
            xi2 = min(max(xi2, 0), N_GRID - 1);
            __builtin_prefetch(&SoS[(size_t)row2 * N_GRID + xi2], 0, 3);
        }

        const int   jc = min(j, N_INT - 1);
        const float si = s_steps[jc];
        const float xs = l * si;
        const float px = x - xs * sth;
        const float py = y - xs * cth;
        int xi = (int)rintf((px - x0) / dx);       // true divide + RNE round = jnp.round
        int yi = (int)rintf((py - y0) / dy);
        const int row = (-yi) & (N_GRID - 1);      // Python negative-row wraparound
        xi = min(max(xi, 0), N_GRID - 1);

        const float sos = SoS[(size_t)row * N_GRID + xi];
        const float v   = 1.0f - v0 / sos;

        const float sm = s_steps[max(jc - 1, 0)];
        const float sp = s_steps[min(jc + 1, N_INT - 1)];
        float w = 0.5f * ((l * si - l * sm) + (l * sp - l * si));
        w = (j < N_INT) ? w : 0.0f;                // single cndmask kills the tail
        return w * v;
    };

    // Two deterministic accumulator chains (nodes tid+512k and tid+256+512k):
    // uniform trip count, no parity selects, two independent gathers in flight.
    // 64 partials per wave = one 16x4 f32 A-matrix for the WMMA reduction.
    v2f acc = {0.0f, 0.0f};
    for (int i = tid; i < N_INT; i += 2 * BLOCK) {
        acc.x += contrib(i);
        acc.y += contrib(i + BLOCK);
    }

    // Wave reduction on the matrix pipe: D = A(16x4 partials) x B(4x16 ones) + 0.
    // Sum of a lane's 8 D regs = half-wave sum (lanes 0-15: rows 0-7;
    // lanes 16-31: rows 8-15); finish with ds_swizzle SWAPX16 (0x401F).
    // EXEC is all-1s here (full block, loops reconverged) as WMMA requires.
    v2f bones = {1.0f, 1.0f};
    v8f c = {};
    c = __builtin_amdgcn_wmma_f32_16x16x4_f32(
            /*neg_a=*/false, acc, /*neg_b=*/false, bones,
            /*c_mod=*/(short)0, c, /*reuse_a=*/false, /*reuse_b=*/false);
    float h = ((c[0] + c[1]) + (c[2] + c[3])) + ((c[4] + c[5]) + (c[6] + c[7]));
    const float other =
        __uint_as_float(__builtin_amdgcn_ds_swizzle(__float_as_uint(h), 0x401F));
    const float wave_sum = h + other;

    const int lane = tid & 31;
    const int wid  = tid >> 5;
    if (lane == 0) s_red[wid] = wave_sum;
    __syncthreads();
    if (tid == 0) {
        float t = 0.0f;
        #pragma unroll
        for (int w = 0; w < BLOCK / 32; ++w) t += s_red[w];
        out[p]            = theta;  // first output: thetas
        out[N_POINTS + p] = t;      // second output: wf
    }
}

extern "C" void kernel_launch(void* const* d_in, const int* in_sizes, int n_in,
                              void* d_out, int out_size, void* d_ws, size_t ws_size,
                              hipStream_t stream) {
    // setup_inputs() order: x, y, SoS, thetas, steps, x0, y0, dx, dy, R_body, v0
    const float* xp     = (const float*)d_in[0];
    const float* yp     = (const float*)d_in[1];
    const float* SoS    = (const float*)d_in[2];
    const float* thetas = (const float*)d_in[3];
    const float* steps  = (const float*)d_in[4];
    const float* x0p    = (const float*)d_in[5];
    const float* y0p    = (const float*)d_in[6];
    const float* dxp    = (const float*)d_in[7];
    const float* dyp    = (const float*)d_in[8];
    const float* Rp     = (const float*)d_in[9];
    const float* v0p    = (const float*)d_in[10];
    float* out = (float*)d_out;

    wavefront_sos_kernel<<<N_POINTS, BLOCK, 0, stream>>>(
        xp, yp, SoS, thetas, steps, x0p, y0p, dxp, dyp, Rp, v0p, out);
}